// DEFNet_45990509805804
// MI455X (gfx1250) — compile-verified
//
#include <hip/hip_runtime.h>

typedef __attribute__((ext_vector_type(2))) float v2f;
typedef __attribute__((ext_vector_type(8))) float v8f;

#define SEQ_L     262144
#define NCH       16
#define MAXK      33
#define KPAD_STEPS 9          // ceil(33/4) -> K padded to 36
#define TILE_OUT  1024        // outputs per workgroup (per batch row)
#define HALO      16
#define LDS_N     (TILE_OUT + 48)   // [t0-16 .. t0+1055], max used idx = 1058

__global__ __launch_bounds__(256)
void conv16_wmma_f32(const float* __restrict__ x,
                     const float* __restrict__ w,
                     const float* __restrict__ bias,
                     float* __restrict__ out)
{
    __shared__ float xs[LDS_N];

    const int tid  = threadIdx.x;
    const int lane = tid & 31;
    const int wave = tid >> 5;

    const int tilesPerRow = SEQ_L / TILE_OUT;           // 256
    const int brow   = blockIdx.x / tilesPerRow;        // batch row
    const int tchunk = blockIdx.x % tilesPerRow;
    const size_t rowBase = (size_t)brow * SEQ_L;
    const int t0 = tchunk * TILE_OUT;

    // ---- stage x window into LDS (zero-padded at row edges) ----
    for (int i = tid; i < LDS_N; i += 256) {
        int gi = t0 - HALO + i;
        float v = 0.0f;
        if (gi >= 0 && gi < SEQ_L) v = x[rowBase + gi];
        xs[i] = v;
    }

    // ---- build masked A fragments (weights) in registers ----
    // A is 16x4 per step: lane holds row m = lane%16; VGPR pair holds
    // K = 4*kk + 2*h + {0,1}, h = lane/16  (ISA 7.12.2 32-bit A layout)
    const int m = lane & 15;
    const int h = lane >> 4;
    v2f afrag[KPAD_STEPS];
#pragma unroll
    for (int kk = 0; kk < KPAD_STEPS; ++kk) {
        const int k0 = kk * 4 + 2 * h;
        float a0 = 0.0f, a1 = 0.0f;
        if (k0 < MAXK) {
            int d = k0 - 16; d = d < 0 ? -d : d;
            if (d <= m + 1) a0 = w[m * MAXK + k0];
        }
        if (k0 + 1 < MAXK) {
            int d = k0 + 1 - 16; d = d < 0 ? -d : d;
            if (d <= m + 1) a1 = w[m * MAXK + k0 + 1];
        }
        afrag[kk].x = a0;
        afrag[kk].y = a1;
    }

    // bias for the 8 C/D rows this lane holds: rows v + 8*h
    float bv[8];
#pragma unroll
    for (int v = 0; v < 8; ++v) bv[v] = bias[v + 8 * h];

    __syncthreads();

    const int nL = lane & 15;   // output column within 16-wide tile

    // ---- each wave: 8 tiles of 16 outputs (wave-uniform control flow) ----
#pragma unroll 1
    for (int j = 0; j < 8; ++j) {
        const int tt = wave * 128 + j * 16;   // tile start, relative to t0
        v8f c = {0.f, 0.f, 0.f, 0.f, 0.f, 0.f, 0.f, 0.f};

#pragma unroll
        for (int kk = 0; kk < KPAD_STEPS; ++kk) {
            // B is 4x16 per step: lane holds col n = lane%16,
            // rows K = 4*kk + 2*h + {0,1}; B[k,n] = x[t0 + n + k - 16]
            const int idx = tt + nL + kk * 4 + 2 * h;
            v2f bfrag;
            bfrag.x = xs[idx];
            bfrag.y = xs[idx + 1];
            c = __builtin_amdgcn_wmma_f32_16x16x4_f32(
                    false, afrag[kk], false, bfrag,
                    (short)0, c, false, false);
        }

        // ---- epilogue: relu(+bias), mean over 16 channels ----
        float s = 0.0f;
#pragma unroll
        for (int v = 0; v < 8; ++v) {
            float y = c[v] + bv[v];
            s += fmaxf(y, 0.0f);
        }
        s += __shfl_xor(s, 16, 32);           // combine lane halves (rows 0-7 + 8-15)
        if (lane < 16) {
            out[rowBase + t0 + tt + nL] = s * (1.0f / 16.0f);
        }
    }
}

extern "C" void kernel_launch(void* const* d_in, const int* in_sizes, int n_in,
                              void* d_out, int out_size, void* d_ws, size_t ws_size,
                              hipStream_t stream) {
    (void)in_sizes; (void)n_in; (void)d_ws; (void)ws_size; (void)out_size;
    const float* x = (const float*)d_in[0];   // [32, 1, 262144] fp32
    const float* w = (const float*)d_in[1];   // [16, 33] fp32
    const float* b = (const float*)d_in[2];   // [16] fp32
    float* out = (float*)d_out;               // [32, 262144] fp32

    const int blocks = 32 * (SEQ_L / TILE_OUT);   // 8192
    conv16_wmma_f32<<<blocks, 256, 0, stream>>>(x, w, b, out);
}